// Attention_21869973471247
// MI455X (gfx1250) — compile-verified
//
#include <hip/hip_runtime.h>
#include <math.h>

typedef float v2f __attribute__((ext_vector_type(2)));
typedef float v8f __attribute__((ext_vector_type(8)));

#define BATCH 4
#define DIM   192
#define IMG_H 128
#define IMG_W 128
#define PIX   16384
#define HEADS 6
#define CPH   32     // channels per head

__device__ __forceinline__ v8f wmma4(v2f a, v2f b, v8f c) {
  // D = A(16x4,f32) * B(4x16,f32) + C(16x16,f32)
  return __builtin_amdgcn_wmma_f32_16x16x4_f32(
      /*neg_a=*/false, a, /*neg_b=*/false, b,
      /*c_mod=*/(short)0, c, /*reuse_a=*/false, /*reuse_b=*/false);
}

// ---------------------------------------------------------------------------
// Generic 1x1-conv GEMM:  Y[b][m][pix] = sum_ic Wm[m][ic] * X[b][ic][pix]
// Wm row-major with row stride DIM. grid = (PIX/128, M/64, BATCH), block 256.
// ---------------------------------------------------------------------------
__global__ __launch_bounds__(256)
void gemm1x1_kernel(const float* __restrict__ Wm, const float* __restrict__ X,
                    float* __restrict__ Y, int M)
{
  const int n0 = blockIdx.x * 128;
  const int m0 = blockIdx.y * 64;
  const int b  = blockIdx.z;
  const float* Xb = X + (size_t)b * DIM * PIX;
  float* Yb = Y + (size_t)b * M * PIX;

  __shared__ float Ws[64][17];       // padded: stride 17 avoids bank conflicts
  __shared__ float Xs[16][128];

  const int tid  = threadIdx.x;
  const int lane = tid & 31;
  const int wm   = (tid >> 5) & 3;   // wave m-subtile (16 rows)
  const int wn   = (tid >> 5) >> 2;  // wave n-half (64 cols)
  const int lrow = lane & 15;
  const int koff = (lane >> 4) * 2;

  v8f acc[4];
  #pragma unroll
  for (int i = 0; i < 4; ++i) acc[i] = (v8f){};

  for (int k0 = 0; k0 < DIM; k0 += 16) {
    // stage W tile: 64 x 16
    {
      int r  = tid >> 2;
      int c4 = (tid & 3) * 4;
      float4 v = *(const float4*)(Wm + (size_t)(m0 + r) * DIM + k0 + c4);
      Ws[r][c4 + 0] = v.x; Ws[r][c4 + 1] = v.y;
      Ws[r][c4 + 2] = v.z; Ws[r][c4 + 3] = v.w;
    }
    // stage X tile: 16 x 128
    {
      int i = tid * 8;
      int r = i >> 7;
      int c = i & 127;
      const float* src = Xb + (size_t)(k0 + r) * PIX + n0 + c;
      *(float4*)&Xs[r][c]     = *(const float4*)(src);
      *(float4*)&Xs[r][c + 4] = *(const float4*)(src + 4);
    }
    if (k0 + 16 < DIM)   // hint next X tile into cache (global_prefetch_b8)
      __builtin_prefetch(Xb + (size_t)(k0 + 16) * PIX + n0 + (tid & 127), 0, 1);
    __syncthreads();

    #pragma unroll
    for (int kk = 0; kk < 16; kk += 4) {
      v2f a;
      a.x = Ws[wm * 16 + lrow][kk + koff];
      a.y = Ws[wm * 16 + lrow][kk + koff + 1];
      #pragma unroll
      for (int nf = 0; nf < 4; ++nf) {
        int col = wn * 64 + nf * 16 + lrow;
        v2f bf;
        bf.x = Xs[kk + koff][col];
        bf.y = Xs[kk + koff + 1][col];
        acc[nf] = wmma4(a, bf, acc[nf]);
      }
    }
    __syncthreads();
  }

  const int rbase = m0 + wm * 16 + (lane >> 4) * 8;
  #pragma unroll
  for (int nf = 0; nf < 4; ++nf) {
    int col = n0 + wn * 64 + nf * 16 + lrow;
    #pragma unroll
    for (int g = 0; g < 8; ++g)
      Yb[(size_t)(rbase + g) * PIX + col] = acc[nf][g];
  }
}

// ---------------------------------------------------------------------------
// Depthwise 3x3, zero-padded. Logical channel ch in [0,576):
//   [0,192)   -> Q from convA row ch      (w_dw row ch)
//   [192,384) -> K from convB row ch-192  (w_dw row ch)
//   [384,576) -> V from convB row ch-192  (w_dw row ch)
// ---------------------------------------------------------------------------
__global__ __launch_bounds__(256)
void dwconv_kernel(const float* __restrict__ convA, const float* __restrict__ convB,
                   const float* __restrict__ w_dw,
                   float* __restrict__ Q, float* __restrict__ K, float* __restrict__ V)
{
  size_t idx = (size_t)blockIdx.x * 256 + threadIdx.x;
  if (idx >= (size_t)BATCH * 576 * PIX) return;
  int pix = (int)(idx % PIX);
  int t   = (int)(idx / PIX);
  int ch  = t % 576;
  int b   = t / 576;
  int y = pix >> 7, x = pix & 127;

  const float* src;
  float* dst;
  if (ch < 192)      { src = convA + ((size_t)b * 192 + ch)        * PIX;
                       dst = Q     + ((size_t)b * 192 + ch)        * PIX; }
  else if (ch < 384) { src = convB + ((size_t)b * 384 + (ch - 192))* PIX;
                       dst = K     + ((size_t)b * 192 + (ch - 192))* PIX; }
  else               { src = convB + ((size_t)b * 384 + (ch - 192))* PIX;
                       dst = V     + ((size_t)b * 192 + (ch - 384))* PIX; }

  const float* wd = w_dw + (size_t)ch * 9;
  float s = 0.f;
  #pragma unroll
  for (int dy = -1; dy <= 1; ++dy) {
    int yy = y + dy;
    if (yy < 0 || yy >= IMG_H) continue;
    #pragma unroll
    for (int dx = -1; dx <= 1; ++dx) {
      int xx = x + dx;
      if (xx < 0 || xx >= IMG_W) continue;
      s += src[yy * IMG_W + xx] * wd[(dy + 1) * 3 + (dx + 1)];
    }
  }
  dst[pix] = s;
}

// ---------------------------------------------------------------------------
// Reciprocal L2 norms of Q / K rows. grid = (768, 2), block 256.
// ---------------------------------------------------------------------------
__global__ __launch_bounds__(256)
void norm_kernel(const float* __restrict__ Q, const float* __restrict__ K,
                 float* __restrict__ rnq, float* __restrict__ rnk)
{
  const float* src = (blockIdx.y == 0 ? Q : K) + (size_t)blockIdx.x * PIX;
  float s = 0.f;
  for (int i = threadIdx.x; i < PIX; i += 256) { float v = src[i]; s += v * v; }
  __shared__ float red[256];
  red[threadIdx.x] = s;
  __syncthreads();
  for (int off = 128; off > 0; off >>= 1) {
    if (threadIdx.x < off) red[threadIdx.x] += red[threadIdx.x + off];
    __syncthreads();
  }
  if (threadIdx.x == 0) {
    float n = fmaxf(sqrtf(red[0]), 1e-12f);
    (blockIdx.y == 0 ? rnq : rnk)[blockIdx.x] = 1.f / n;
  }
}

// ---------------------------------------------------------------------------
// attn[b,h] (32x32) = (q/|q|) (k/|k|)^T * temperature[h]; K=16384 split over
// 8 waves, WMMA partials, LDS ds_add_f32 reduction. grid = 24, block 256.
// ---------------------------------------------------------------------------
__global__ __launch_bounds__(256)
void attn_kernel(const float* __restrict__ Q, const float* __restrict__ K,
                 const float* __restrict__ rnq, const float* __restrict__ rnk,
                 const float* __restrict__ temperature, float* __restrict__ attn)
{
  const int bh = blockIdx.x;
  const int b = bh / HEADS, h = bh % HEADS;
  const float* Qb = Q + ((size_t)b * DIM + h * CPH) * PIX;
  const float* Kb = K + ((size_t)b * DIM + h * CPH) * PIX;

  const int lane = threadIdx.x & 31;
  const int wave = threadIdx.x >> 5;
  const int lrow = lane & 15;
  const int koff = (lane >> 4) * 2;
  const int kbeg = wave * (PIX / 8);
  const int kend = kbeg + (PIX / 8);

  v8f a00 = {}, a01 = {}, a10 = {}, a11 = {};
  for (int kk = kbeg; kk < kend; kk += 4) {
    float2 t;
    v2f qa0, qa1, kb0, kb1;
    t = *(const float2*)(Qb + (size_t)lrow        * PIX + kk + koff); qa0.x = t.x; qa0.y = t.y;
    t = *(const float2*)(Qb + (size_t)(16 + lrow) * PIX + kk + koff); qa1.x = t.x; qa1.y = t.y;
    t = *(const float2*)(Kb + (size_t)lrow        * PIX + kk + koff); kb0.x = t.x; kb0.y = t.y;
    t = *(const float2*)(Kb + (size_t)(16 + lrow) * PIX + kk + koff); kb1.x = t.x; kb1.y = t.y;
    a00 = wmma4(qa0, kb0, a00);
    a01 = wmma4(qa0, kb1, a01);
    a10 = wmma4(qa1, kb0, a10);
    a11 = wmma4(qa1, kb1, a11);
  }

  __shared__ float s_attn[32 * 32];
  for (int i = threadIdx.x; i < 1024; i += 256) s_attn[i] = 0.f;
  __syncthreads();

  const int rb = (lane >> 4) * 8;
  #pragma unroll
  for (int g = 0; g < 8; ++g) {
    atomicAdd(&s_attn[(rb + g)      * 32 + lrow],      a00[g]);
    atomicAdd(&s_attn[(rb + g)      * 32 + 16 + lrow], a01[g]);
    atomicAdd(&s_attn[(16 + rb + g) * 32 + lrow],      a10[g]);
    atomicAdd(&s_attn[(16 + rb + g) * 32 + 16 + lrow], a11[g]);
  }
  __syncthreads();

  const float temp = temperature[h];
  for (int i = threadIdx.x; i < 1024; i += 256) {
    int c = i >> 5, d = i & 31;
    attn[(size_t)bh * 1024 + i] =
        s_attn[i] * rnq[b * DIM + h * CPH + c] * rnk[b * DIM + h * CPH + d] * temp;
  }
}

// ---------------------------------------------------------------------------
// Per-row top-k masked softmax mix: amix = sum_i attn_w[i]*softmax(mask_i(attn)).
// grid = 24, block 32 (one thread per attn row).
// ---------------------------------------------------------------------------
__global__ void amix_kernel(const float* __restrict__ attn,
                            const float* __restrict__ attn_w,
                            float* __restrict__ amix)
{
  const int bh = blockIdx.x;
  const int c  = threadIdx.x;
  const float* row = attn + (size_t)bh * 1024 + c * 32;

  float a[32], s[32];
  for (int d = 0; d < 32; ++d) { a[d] = row[d]; s[d] = a[d]; }
  // insertion sort descending
  for (int i = 1; i < 32; ++i) {
    float key = s[i];
    int j = i - 1;
    while (j >= 0 && s[j] < key) { s[j + 1] = s[j]; --j; }
    s[j + 1] = key;
  }
  const int ks[4] = {16, 21, 24, 25};   // int(C/2), int(2C/3), int(3C/4), int(4C/5)
  const float m = s[0];
  float e[32], out[32];
  for (int d = 0; d < 32; ++d) { e[d] = expf(a[d] - m); out[d] = 0.f; }
  for (int i = 0; i < 4; ++i) {
    float thr = s[ks[i] - 1];
    float Z = 0.f;
    for (int d = 0; d < 32; ++d) if (a[d] >= thr) Z += e[d];
    float wz = attn_w[i] / Z;
    for (int d = 0; d < 32; ++d) if (a[d] >= thr) out[d] += wz * e[d];
  }
  float* dst = amix + (size_t)bh * 1024 + c * 32;
  for (int d = 0; d < 32; ++d) dst[d] = out[d];
}

// ---------------------------------------------------------------------------
// out_pre[b, h*32+c, pix] = sum_d amix[b,h][c][d] * V[b, h*32+d, pix]
// grid = (PIX/128, 24), block 256. amix staged in LDS, V loaded coalesced.
// ---------------------------------------------------------------------------
__global__ __launch_bounds__(256)
void av_kernel(const float* __restrict__ amix, const float* __restrict__ V,
               float* __restrict__ outp)
{
  const int bh = blockIdx.y;
  const int b = bh / HEADS, h = bh % HEADS;
  const int n0 = blockIdx.x * 128;

  __shared__ float s_a[1024];
  for (int i = threadIdx.x; i < 1024; i += 256)
    s_a[i] = amix[(size_t)bh * 1024 + i];
  __syncthreads();

  const int lane = threadIdx.x & 31;
  const int wave = threadIdx.x >> 5;
  const int lrow = lane & 15;
  const int koff = (lane >> 4) * 2;
  const int col  = n0 + wave * 16 + lrow;

  const float* Vh = V + ((size_t)b * DIM + h * CPH) * PIX;
  v8f acc0 = {}, acc1 = {};
  #pragma unroll
  for (int kk = 0; kk < 32; kk += 4) {
    v2f bf, a0, a1;
    bf.x = Vh[(size_t)(kk + koff)     * PIX + col];
    bf.y = Vh[(size_t)(kk + koff + 1) * PIX + col];
    a0.x = s_a[lrow        * 32 + kk + koff];
    a0.y = s_a[lrow        * 32 + kk + koff + 1];
    a1.x = s_a[(16 + lrow) * 32 + kk + koff];
    a1.y = s_a[(16 + lrow) * 32 + kk + koff + 1];
    acc0 = wmma4(a0, bf, acc0);
    acc1 = wmma4(a1, bf, acc1);
  }

  const int rb = (lane >> 4) * 8;
  float* dst = outp + ((size_t)b * DIM + h * CPH) * PIX;
  #pragma unroll
  for (int g = 0; g < 8; ++g) {
    dst[(size_t)(rb + g)      * PIX + col] = acc0[g];
    dst[(size_t)(16 + rb + g) * PIX + col] = acc1[g];
  }
}

// ---------------------------------------------------------------------------
extern "C" void kernel_launch(void* const* d_in, const int* in_sizes, int n_in,
                              void* d_out, int out_size, void* d_ws, size_t ws_size,
                              hipStream_t stream)
{
  const float* x      = (const float*)d_in[0];   // [4,192,128,128]
  const float* ref    = (const float*)d_in[1];   // [4,192,128,128]
  const float* w_qkv  = (const float*)d_in[2];   // [576,192]
  const float* w_dw   = (const float*)d_in[3];   // [576,9]
  const float* w_proj = (const float*)d_in[4];   // [192,192]
  const float* temp   = (const float*)d_in[5];   // [6]
  const float* attn_w = (const float*)d_in[6];   // [4]
  float* out = (float*)d_out;
  float* ws  = (float*)d_ws;

  const size_t szA = (size_t)BATCH * 192 * PIX;  // conv1x1(x) q-channels
  const size_t szB = (size_t)BATCH * 384 * PIX;  // conv1x1(ref) kv-channels
  const size_t szQ = (size_t)BATCH * 192 * PIX;

  float* convA  = ws;                       // also reused as out_pre
  float* convB  = convA + szA;
  float* Qb     = convB + szB;
  float* Kb     = Qb + szQ;
  float* Vb     = Kb + szQ;
  float* rnq    = Vb + szQ;
  float* rnk    = rnq + BATCH * DIM;
  float* attn   = rnk + BATCH * DIM;        // 24 * 1024
  float* amix   = attn + 24 * 1024;
  float* outpre = convA;                    // reuse (convA dead after dwconv)

  dim3 blk(256);

  // conv1x1: q-part of qkv(x), kv-part of qkv(ref)
  gemm1x1_kernel<<<dim3(PIX / 128, 192 / 64, BATCH), blk, 0, stream>>>(
      w_qkv, x, convA, 192);
  gemm1x1_kernel<<<dim3(PIX / 128, 384 / 64, BATCH), blk, 0, stream>>>(
      w_qkv + (size_t)192 * DIM, ref, convB, 384);

  // depthwise 3x3 -> Q, K, V
  dwconv_kernel<<<(BATCH * 576 * PIX) / 256, blk, 0, stream>>>(
      convA, convB, w_dw, Qb, Kb, Vb);

  // reciprocal L2 norms
  norm_kernel<<<dim3(BATCH * DIM, 2), blk, 0, stream>>>(Qb, Kb, rnq, rnk);

  // channel attention logits
  attn_kernel<<<BATCH * HEADS, blk, 0, stream>>>(Qb, Kb, rnq, rnk, temp, attn);

  // top-k masked softmax mixture
  amix_kernel<<<BATCH * HEADS, 32, 0, stream>>>(attn, attn_w, amix);

  // attention * V
  av_kernel<<<dim3(PIX / 128, BATCH * HEADS), blk, 0, stream>>>(amix, Vb, outpre);

  // final projection
  gemm1x1_kernel<<<dim3(PIX / 128, 192 / 64, BATCH), blk, 0, stream>>>(
      w_proj, outpre, out, 192);
}